// Splats2D_40638980555165
// MI455X (gfx1250) — compile-verified
//
#include <hip/hip_runtime.h>
#include <math.h>

// ---------------------------------------------------------------------------
// 2D Gaussian splat rasteriser for MI455X (gfx1250, wave32).
//
//  q(p,g) = -0.5*log2e * mahalanobis(p,g) is bilinear in pixel coords, so it
//  is computed as a 16x16x4 f32 WMMA:  A = per-pixel features [lx^2, lx, 1, 0]
//  (tile-local x), B = per-gaussian coefficients [c0, c3', c5', 0] with the
//  tile origin folded into c3'/c5'.  density = exp2(q).
//  Front-to-back compositing over depth-sorted gaussians is done per pixel
//  after transposing the WMMA result tile through LDS, with wave-uniform
//  early exit once the whole tile's transmittance has saturated.
// ---------------------------------------------------------------------------

typedef __attribute__((ext_vector_type(2))) float v2f;
typedef __attribute__((ext_vector_type(8))) float v8f;

#define G_N   1024
#define IMG   384
#define KNEG  (-0.72134752044448170f)   // -0.5 * log2(e)

// ---------------------------------------------------------------------------
// Pre-pass: covariance inverse, sigmoids, stable depth sort (rank via O(G^2)
// comparison against LDS copy of depths).  Writes depth-sorted records:
//   geo[rank*8 + 0..4] = kA, kB, kC, mx, my      (kA=k*iA, kB=2k*iB, kC=k*iC)
//   col[rank]          = (r, g, b, opacity)       (all sigmoided)
// ---------------------------------------------------------------------------
__global__ __launch_bounds__(G_N) void splat_prep(
    const float* __restrict__ means,          // (G,2)
    const float* __restrict__ log_scale,      // (G,2)
    const float* __restrict__ rot,            // (G,)
    const float* __restrict__ colour_logits,  // (G,3)
    const float* __restrict__ opacity_logit,  // (G,)
    const float* __restrict__ depth,          // (G,)
    float*       __restrict__ geo,            // (G,8)
    float4*      __restrict__ col)            // (G,)
{
    __shared__ float sd[G_N];
    const int g = threadIdx.x;

    const float d = depth[g];
    sd[g] = d;
    __syncthreads();

    int rank = 0;
    for (int j = 0; j < G_N; ++j) {
        const float dj = sd[j];
        rank += (dj < d) || (dj == d && j < g);   // stable argsort rank
    }

    const float s2x = __expf(2.0f * log_scale[2 * g + 0]);
    const float s2y = __expf(2.0f * log_scale[2 * g + 1]);
    const float c   = __cosf(rot[g]);
    const float si  = __sinf(rot[g]);

    const float cov_a = c * c * s2x + si * si * s2y;
    const float cov_b = c * si * (s2x - s2y);
    const float cov_d = si * si * s2x + c * c * s2y;
    const float det   = cov_a * cov_d - cov_b * cov_b;
    const float iA    =  cov_d / det;
    const float iB    = -cov_b / det;
    const float iC    =  cov_a / det;

    geo[8 * rank + 0] = KNEG * iA;
    geo[8 * rank + 1] = KNEG * 2.0f * iB;
    geo[8 * rank + 2] = KNEG * iC;
    geo[8 * rank + 3] = means[2 * g + 0];
    geo[8 * rank + 4] = means[2 * g + 1];
    geo[8 * rank + 5] = 0.0f;
    geo[8 * rank + 6] = 0.0f;
    geo[8 * rank + 7] = 0.0f;

    #define SIGM(x) (1.0f / (1.0f + __builtin_amdgcn_exp2f(-(x) * 1.4426950408889634f)))
    float4 c4;
    c4.x = SIGM(colour_logits[3 * g + 0]);
    c4.y = SIGM(colour_logits[3 * g + 1]);
    c4.z = SIGM(colour_logits[3 * g + 2]);
    c4.w = SIGM(opacity_logit[g]);
    #undef SIGM
    col[rank] = c4;
}

// ---------------------------------------------------------------------------
// Rasteriser: one wave32 per 16(x) x 2(y) pixel tile.
// Per 16-gaussian chunk: 2x v_wmma_f32_16x16x4_f32 (one per image row),
// transpose D through LDS (row stride 19 => conflict-free composite reads),
// then per-pixel sequential alpha compositing (all 32 lanes active).
// ---------------------------------------------------------------------------
__global__ __launch_bounds__(32) void splat_raster(
    const float*  __restrict__ geo,
    const float4* __restrict__ col,
    float*        __restrict__ out)
{
    const int lane = threadIdx.x;          // 0..31
    const int n    = lane & 15;            // gaussian slot / pixel x in tile
    const int half = lane >> 4;            // 0 or 1

    const int tx = blockIdx.x % (IMG / 16);
    const int ty = blockIdx.x / (IMG / 16);
    const float x0  = (float)(tx * 16);
    const float y0f = (float)(ty * 2);

    __shared__ float  dt[2 * 16 * 19];     // two 16x16 q-tiles, row stride 19
    __shared__ float4 cs[16];              // chunk colours+opacity

    // A matrix (16 pixels x 4 features), constant for the wave:
    // lanes 0-15 : VGPR0 = K0 = lx^2 , VGPR1 = K1 = lx     (row M = lane)
    // lanes 16-31: VGPR0 = K2 = 1    , VGPR1 = K3 = 0
    const float lxf = (float)n;
    v2f A;
    A.x = half ? 1.0f : lxf * lxf;
    A.y = half ? 0.0f : lxf;

    float T = 1.0f, cr = 0.0f, cg = 0.0f, cb = 0.0f;

    for (int chunk = 0; chunk < G_N / 16; ++chunk) {
        const int gi = chunk * 16 + n;
        const float4* gp = (const float4*)(geo + 8 * gi);
        const float4 g0 = gp[0];           // kA kB kC mx
        const float4 g1 = gp[1];           // my  -  -  -
        const float kA = g0.x, kB = g0.y, kC = g0.z;
        const float mxp = g0.w - x0;       // tile-local mean x
        const float my0 = g1.x - y0f;      // tile-local mean y, row 0

        const float c3_0 = -2.0f * kA * mxp - kB * my0;
        const float c5_0 = kA * mxp * mxp + kB * mxp * my0 + kC * my0 * my0;
        // Exact incremental update for row 1 (my1 = my0 - 1):
        const float c3_1 = c3_0 + kB;
        const float c5_1 = c5_0 - kB * mxp - kC * (2.0f * my0 - 1.0f);

        // B matrix (4 features x 16 gaussians):
        // lanes 0-15 : VGPR0 = row K0 = c0 , VGPR1 = row K1 = c3   (N = lane)
        // lanes 16-31: VGPR0 = row K2 = c5 , VGPR1 = row K3 = 0
        v2f B0, B1;
        B0.x = half ? c5_0 : kA;
        B0.y = half ? 0.0f : c3_0;
        B1.x = half ? c5_1 : kA;
        B1.y = half ? 0.0f : c3_1;

        v8f Z = {};
        v8f D0 = __builtin_amdgcn_wmma_f32_16x16x4_f32(
            false, A, false, B0, (short)0, Z, false, false);
        v8f D1 = __builtin_amdgcn_wmma_f32_16x16x4_f32(
            false, A, false, B1, (short)0, Z, false, false);

        // Scatter D into LDS transposed.
        // D layout: VGPR r, lanes 0-15 -> M=r ; lanes 16-31 -> M=r+8.
        #pragma unroll
        for (int r = 0; r < 8; ++r) {
            const int x = half * 8 + r;    // pixel x within tile
            dt[x * 19 + n]       = D0[r];  // image row y0
            dt[304 + x * 19 + n] = D1[r];  // image row y0+1
        }
        if (half == 0) cs[n] = col[gi];
        __syncthreads();

        // Front-to-back composite: lane handles pixel (x = n, row = half).
        const int base = half * 304 + n * 19;
        #pragma unroll
        for (int j = 0; j < 16; ++j) {
            const float  q  = dt[base + j];
            const float4 c4 = cs[j];                     // LDS broadcast
            const float  a  = fminf(c4.w * __builtin_amdgcn_exp2f(q), 0.99f);
            const float  w  = T * a;
            cr += w * c4.x;
            cg += w * c4.y;
            cb += w * c4.z;
            T  -= w;                                     // T *= (1 - a)
        }
        __syncthreads();

        // Wave-uniform early exit: every remaining contribution is < 1e-6.
        // Ballot is uniform across the wave, so EXEC stays all-ones for the
        // WMMAs on the path that continues.
        if (__builtin_amdgcn_ballot_w32(T > 1.0e-6f) == 0u) break;
    }

    const int x = tx * 16 + n;
    const int y = ty * 2 + half;
    float* o = out + (size_t)(y * IMG + x) * 3;
    o[0] = cr;
    o[1] = cg;
    o[2] = cb;
}

// ---------------------------------------------------------------------------
extern "C" void kernel_launch(void* const* d_in, const int* in_sizes, int n_in,
                              void* d_out, int out_size, void* d_ws, size_t ws_size,
                              hipStream_t stream)
{
    (void)in_sizes; (void)n_in; (void)out_size; (void)ws_size;

    const float* means         = (const float*)d_in[0];
    const float* log_scale     = (const float*)d_in[1];
    const float* rot           = (const float*)d_in[2];
    const float* colour_logits = (const float*)d_in[3];
    const float* opacity_logit = (const float*)d_in[4];
    const float* depth         = (const float*)d_in[5];
    // d_in[6] = image_size scalar (384), compile-time constant here.

    float*  geo = (float*)d_ws;                                     // G*8 f32
    float4* col = (float4*)((char*)d_ws + G_N * 8 * sizeof(float)); // G float4
    float*  out = (float*)d_out;

    splat_prep<<<1, G_N, 0, stream>>>(means, log_scale, rot, colour_logits,
                                      opacity_logit, depth, geo, col);

    const int tiles = (IMG / 16) * (IMG / 2);  // 24 * 192 = 4608 waves
    splat_raster<<<tiles, 32, 0, stream>>>(geo, col, out);
}